// ToggleSwitch_25881472926367
// MI455X (gfx1250) — compile-verified
//
#include <hip/hip_runtime.h>
#include <math.h>
#include <float.h>

// Problem constants from the reference
#define NN     4096
#define MM     64
#define TT     300
#define CHUNK  30                 // timesteps staged per LDS pass (300 = 10 * 30)
#define BLOCK  256                // 8 wave32 waves per block
#define ROWF   (2 * CHUNK + 2)    // 62 floats per pair row; pad -> conflict-free LDS banks
#define NPAIR  (NN * MM)

#define HAS_ASYNC_LDS __has_builtin(__builtin_amdgcn_global_load_async_to_lds_b64)

typedef int v2i_t __attribute__((ext_vector_type(2)));

// -------- fast device helpers ----------------------------------------------

// x >= 0 guaranteed by the truncated-normal construction.
// pow(x, e) = exp2(e * log2(x)); log2(0) = -inf -> exp2(-inf) = 0, matching pow(0,e>0)=0.
__device__ __forceinline__ float pow_nonneg(float x, float e) {
#if __has_builtin(__builtin_amdgcn_exp2f) && __has_builtin(__builtin_amdgcn_logf)
    return __builtin_amdgcn_exp2f(e * __builtin_amdgcn_logf(x));   // v_exp_f32(e * v_log_f32(x))
#else
    return exp2f(e * log2f(x));
#endif
}

__device__ __forceinline__ float fast_rcp(float x) {
#if __has_builtin(__builtin_amdgcn_rcpf)
    return __builtin_amdgcn_rcpf(x);
#else
    return 1.0f / x;
#endif
}

// _truncated_normal(noise, loc, std):
//   phi_a = erf(-loc/std); return erfinv(phi_a + noise*(1-phi_a))*std + loc
__device__ __forceinline__ float trunc_norm(float noise, float loc, float std_, float inv_std) {
    float phi_a = erff(-loc * inv_std);
    float r     = fmaf(noise, 1.0f - phi_a, phi_a);
    return fmaf(erfinvf(r), std_, loc);
}

// -------- kernel ------------------------------------------------------------

__global__ __launch_bounds__(BLOCK)
void toggle_switch_kernel(const float* __restrict__ theta,
                          const float* __restrict__ noise1,   // (N, M, T, 2)
                          const float* __restrict__ noise2,   // (N, M, 1)
                          float* __restrict__ out)            // (N, M)
{
    __shared__ float lds[BLOCK * ROWF];   // 62 KB

    const int tid      = threadIdx.x;
    const int pairBase = blockIdx.x * BLOCK;
    const int pair     = pairBase + tid;  // n*M + m
    const int n        = pair >> 6;       // M == 64

    // Per-trajectory parameters (64 threads share each theta row; L0/L2 broadcast)
    const float a1 = theta[n * 7 + 0];
    const float a2 = theta[n * 7 + 1];
    const float b1 = theta[n * 7 + 2];
    const float b2 = theta[n * 7 + 3];
    const float mu = theta[n * 7 + 4];
    const float sg = theta[n * 7 + 5];
    const float gm = theta[n * 7 + 6];

    const float kappa    = 1.0f;
    const float delta    = 0.03f;
    const float step_std = 0.5f;
    const float inv_sstd = 2.0f;

    float u = 10.0f, v = 10.0f;

    const float2* __restrict__ noise1v = (const float2*)noise1;  // index: pair*T + t

    for (int c = 0; c < TT / CHUNK; ++c) {
        // ---- stage chunk c into LDS, coalesced: 240B contiguous runs per pair ----
        #pragma unroll
        for (int it = 0; it < CHUNK; ++it) {
            unsigned i  = (unsigned)(it * BLOCK + tid);
            unsigned p  = i / (unsigned)CHUNK;       // pair-in-block
            unsigned t2 = i - p * (unsigned)CHUNK;   // timestep-in-chunk
            const float2* src = noise1v + (size_t)(pairBase + p) * TT + (size_t)(c * CHUNK + t2);
            float* dst = &lds[p * ROWF + 2u * t2];
#if HAS_ASYNC_LDS
            // generic -> typed generic first, then AS-only casts (lower to addrspacecast)
            v2i_t* snc = (v2i_t*)const_cast<float2*>(src);
            v2i_t* dnc = (v2i_t*)dst;
            __builtin_amdgcn_global_load_async_to_lds_b64(
                (__attribute__((address_space(1))) v2i_t*)snc,
                (__attribute__((address_space(3))) v2i_t*)dnc,
                0, 0);
#else
            *(float2*)dst = *src;
#endif
        }
#if HAS_ASYNC_LDS
        asm volatile("s_wait_asynccnt 0" ::: "memory");
#endif
        __syncthreads();

        // ---- consume this thread's private LDS row: 30 scan steps ----
        const float* __restrict__ row = &lds[tid * ROWF];
        #pragma unroll 2
        for (int t2 = 0; t2 < CHUNK; ++t2) {
            const float nu = row[2 * t2 + 0];
            const float nv = row[2 * t2 + 1];

            const float u_old = u;
            // u_mean = u + a1/(1 + v^b1) - (kappa + delta*u)
            float pv     = pow_nonneg(v, b1);
            float u_mean = fmaf(-delta, u, u - kappa) + a1 * fast_rcp(1.0f + pv);
            u = trunc_norm(nu, u_mean, step_std, inv_sstd);

            // v_mean = v + a2/(1 + u_old^b2) - (kappa + delta*v)   (uses OLD u)
            float pu     = pow_nonneg(u_old, b2);
            float v_mean = fmaf(-delta, v, v - kappa) + a2 * fast_rcp(1.0f + pu);
            v = trunc_norm(nv, v_mean, step_std, inv_sstd);
        }
        __syncthreads();   // LDS reused by next chunk
    }

    // ---- observation model ----
    // x = trunc_norm(noise2, mu + u_T, mu*sigma / u_T^gamma); nan->0; min(x, 2000)
    const float loc  = mu + u;
    const float ug   = pow_nonneg(u, gm);
    const float std_ = (mu * sg) / ug;            // exact div: preserve inf semantics at u==0
    const float phia = erff(-loc / std_);
    const float r    = fmaf(noise2[pair], 1.0f - phia, phia);
    float x = fmaf(erfinvf(r), std_, loc);

    if (__builtin_isnan(x)) {
        x = 0.0f;                                  // jnp.nan_to_num(nan=0.0)
    } else if (__builtin_isinf(x)) {
        x = (x > 0.0f) ? FLT_MAX : -FLT_MAX;       // nan_to_num default +/-inf mapping
    }
    x = fminf(x, 2000.0f);

    out[pair] = x;
}

// -------- launch ------------------------------------------------------------

extern "C" void kernel_launch(void* const* d_in, const int* in_sizes, int n_in,
                              void* d_out, int out_size, void* d_ws, size_t ws_size,
                              hipStream_t stream) {
    const float* theta  = (const float*)d_in[0];   // (4096, 7)
    const float* noise1 = (const float*)d_in[1];   // (4096, 64, 300, 2)
    const float* noise2 = (const float*)d_in[2];   // (4096, 64, 1)
    float*       out    = (float*)d_out;           // (4096, 64)

    (void)in_sizes; (void)n_in; (void)out_size; (void)d_ws; (void)ws_size;

    dim3 grid(NPAIR / BLOCK);   // 1024 blocks
    dim3 block(BLOCK);
    toggle_switch_kernel<<<grid, block, 0, stream>>>(theta, noise1, noise2, out);
}